// RandomForest_52175262712157
// MI455X (gfx1250) — compile-verified
//
#include <hip/hip_runtime.h>
#include <hip/hip_bf16.h>

// ---------------------------------------------------------------------------
// Random forest inference on MI455X:
//   (1) TDM (tensor_load_to_lds) stages the 16x256 f32 node-weight tile of one
//       tree into LDS (once per block, padded rows -> conflict-free ds_load_b64),
//   (2) dense f32 tensor-core GEMM S[n, b] = W_t . x^T via V_WMMA_F32_16X16X4_F32
//       with register double-buffered B fragments (loads overlap WMMAs),
//   (3) per-sample 5-level routing on the LDS score tile,
//   (4) leaf gather + mean over trees.
// ---------------------------------------------------------------------------

#define NUM_TREES   100
#define NUM_NODES   31      // stored per tree; only 0..15 are reachable
#define USED_NODES  16
#define INPUT_DIM   256
#define BATCH       8192
#define NUM_LEAVES  32
#define NUM_CLASSES 64
#define TREE_DEPTH  5

#define W_LDS_STRIDE 260    // 256 + 4 dwords TDM pad -> stride%64 == 4 (no bank conflicts)

typedef __attribute__((ext_vector_type(2))) float v2f;
typedef __attribute__((ext_vector_type(8))) float v8f;
typedef __attribute__((ext_vector_type(4))) unsigned int u32x4;
typedef __attribute__((ext_vector_type(4))) int i32x4;
typedef __attribute__((ext_vector_type(8))) int i32x8;

// Each block: one tree, 512 samples. 8 waves; each wave produces a
// 16(node) x 64(sample) fp32 score tile via 64 K-steps of WMMA 16x16x4.
__global__ __launch_bounds__(256) void forest_scores_kernel(
    const float* __restrict__ x,          // [BATCH, INPUT_DIM]
    const float* __restrict__ node_w,     // [T, NUM_NODES, INPUT_DIM]
    const float* __restrict__ node_b,     // [T, NUM_NODES]
    unsigned char* __restrict__ leaf_idx) // [T, BATCH]
{
    __shared__ float sW[USED_NODES][W_LDS_STRIDE]; // 16.25 KB, TDM-staged, padded
    __shared__ float sS[8][USED_NODES][64];        // 32 KB: per-wave score tiles

    const int t    = blockIdx.y;
    const int wave = threadIdx.x >> 5;
    const int lane = threadIdx.x & 31;
    const int m    = lane & 15;     // A row (node) / B column within subtile
    const int h    = lane >> 4;     // half-wave selector for K pairs
    const int k2   = 2 * h;
    const int colBase = blockIdx.x * 512 + wave * 64;  // first sample column

    // ---- TDM: stage W_t[0:16, 0:256] into LDS with row padding (wave 0 only).
    // D# per CDNA5 ISA. 2D tensor, 4-byte elems,
    // pad_interval=7 (256 dwords) + pad_amount=3 (4 dwords) => LDS row stride 260.
    if (threadIdx.x < 32) {
        const unsigned long long gaddr =
            (unsigned long long)(const void*)(node_w + (size_t)t * NUM_NODES * INPUT_DIM);
        const unsigned int lds_addr = (unsigned int)(unsigned long long)(const void*)&sW[0][0];

        u32x4 g0;
        g0[0] = 1u;                                    // count=1 (valid user D#)
        g0[1] = lds_addr;                              // lds_addr (bytes)
        g0[2] = (unsigned int)gaddr;                   // global_addr[31:0]
        g0[3] = (unsigned int)(gaddr >> 32) | (2u << 30); // addr[56:32] | type=2

        i32x8 g1;
        g1[0] = (int)((2u << 16)                       // data_size = 4B
                    | (1u << 20)                       // pad_enable
                    | (7u << 22)                       // pad_interval: 256 dwords
                    | (3u << 25));                     // pad_amount: 4 dwords
        g1[1] = (int)((INPUT_DIM & 0xFFFFu) << 16);    // tensor_dim0 lo16
        g1[2] = (int)((INPUT_DIM >> 16) | (USED_NODES << 16)); // dim0 hi | dim1 lo
        g1[3] = (int)(INPUT_DIM << 16);                // dim1 hi(0) | tile_dim0
        g1[4] = USED_NODES;                            // tile_dim1 (tile_dim2=0)
        g1[5] = INPUT_DIM;                             // tensor_dim0_stride lo32
        g1[6] = 0;
        g1[7] = 0;

        i32x4 gz4 = {0, 0, 0, 0};                      // groups 2/3 unused (2D)
        i32x8 gz8 = {0, 0, 0, 0, 0, 0, 0, 0};
        __builtin_amdgcn_tensor_load_to_lds(g0, g1, gz4, gz4, gz8, 0);
        __builtin_amdgcn_s_wait_tensorcnt(0);
    }
    __syncthreads();

    // 4 sample-column subtiles of 16, handled by this wave
    const float* __restrict__ x0 = x + (size_t)(colBase +  0 + m) * INPUT_DIM + k2;
    const float* __restrict__ x1 = x + (size_t)(colBase + 16 + m) * INPUT_DIM + k2;
    const float* __restrict__ x2 = x + (size_t)(colBase + 32 + m) * INPUT_DIM + k2;
    const float* __restrict__ x3 = x + (size_t)(colBase + 48 + m) * INPUT_DIM + k2;
    const float* __restrict__ wm = &sW[m][k2];         // LDS, conflict-free b64

    v8f acc0 = {}, acc1 = {}, acc2 = {}, acc3 = {};

    // ISA layout for V_WMMA_F32_16X16X4_F32: lane l holds K = {2h, 2h+1}
    // in its VGPR pair -> consecutive float2 fetches for both A and B.
    // Software pipeline: B fragments for step kk+4 are in flight while the
    // four WMMAs for step kk execute.
    v2f b0c = *(const v2f*)(x0);
    v2f b1c = *(const v2f*)(x1);
    v2f b2c = *(const v2f*)(x2);
    v2f b3c = *(const v2f*)(x3);

#pragma unroll 4
    for (int kk = 0; kk < INPUT_DIM - 4; kk += 4) {
        v2f b0n = *(const v2f*)(x0 + kk + 4);
        v2f b1n = *(const v2f*)(x1 + kk + 4);
        v2f b2n = *(const v2f*)(x2 + kk + 4);
        v2f b3n = *(const v2f*)(x3 + kk + 4);
        v2f a   = *(const v2f*)(wm + kk);              // ds_load_b64
        acc0 = __builtin_amdgcn_wmma_f32_16x16x4_f32(false, a, false, b0c, (short)0, acc0, false, false);
        acc1 = __builtin_amdgcn_wmma_f32_16x16x4_f32(false, a, false, b1c, (short)0, acc1, false, false);
        acc2 = __builtin_amdgcn_wmma_f32_16x16x4_f32(false, a, false, b2c, (short)0, acc2, false, false);
        acc3 = __builtin_amdgcn_wmma_f32_16x16x4_f32(false, a, false, b3c, (short)0, acc3, false, false);
        b0c = b0n; b1c = b1n; b2c = b2n; b3c = b3n;
    }
    {
        v2f a = *(const v2f*)(wm + INPUT_DIM - 4);
        acc0 = __builtin_amdgcn_wmma_f32_16x16x4_f32(false, a, false, b0c, (short)0, acc0, false, false);
        acc1 = __builtin_amdgcn_wmma_f32_16x16x4_f32(false, a, false, b1c, (short)0, acc1, false, false);
        acc2 = __builtin_amdgcn_wmma_f32_16x16x4_f32(false, a, false, b2c, (short)0, acc2, false, false);
        acc3 = __builtin_amdgcn_wmma_f32_16x16x4_f32(false, a, false, b3c, (short)0, acc3, false, false);
    }

    // Bias add + spill 16x64 tile to LDS.
    // C/D layout: VGPR r holds row (r + 8h), column m.
    const float* __restrict__ nb = node_b + (size_t)t * NUM_NODES;
#pragma unroll
    for (int r = 0; r < 8; ++r) {
        const int node = r + 8 * h;
        const float bias = nb[node];
        sS[wave][node][ 0 + m] = acc0[r] + bias;
        sS[wave][node][16 + m] = acc1[r] + bias;
        sS[wave][node][32 + m] = acc2[r] + bias;
        sS[wave][node][48 + m] = acc3[r] + bias;
    }
    __syncthreads();

    // Routing: 512 samples per block, 2 per thread.
    // bit = (sigmoid(z) <= 0.5) == (z <= 0); idx = 2*idx + bit.
#pragma unroll
    for (int p = 0; p < 2; ++p) {
        const int cl = threadIdx.x * 2 + p;      // 0..511
        const int wv = cl >> 6;
        const int cc = cl & 63;
        const float* __restrict__ S = &sS[wv][0][cc];   // stride 64 per node
        int idx = 0;
#pragma unroll
        for (int lvl = 0; lvl < TREE_DEPTH; ++lvl) {
            const float z = S[idx * 64];                 // idx stays in [0,16)
            idx = 2 * idx + ((z <= 0.0f) ? 1 : 0);
        }
        const int gcol = blockIdx.x * 512 + cl;
        leaf_idx[(size_t)t * BATCH + gcol] = (unsigned char)idx;
    }
}

// Leaf gather + mean over trees. One 64-thread group per sample (class = thread),
// leaves table (0.8 MB) is cache-resident; loads are coalesced across classes.
__global__ __launch_bounds__(256) void forest_avg_kernel(
    const float* __restrict__ leaves,          // [T, NUM_LEAVES, NUM_CLASSES]
    const unsigned char* __restrict__ leaf_idx,// [T, BATCH]
    float* __restrict__ out)                   // [BATCH, NUM_CLASSES]
{
    const int b = blockIdx.x * 4 + (threadIdx.x >> 6);
    const int c = threadIdx.x & 63;
    float acc = 0.0f;
#pragma unroll 4
    for (int t = 0; t < NUM_TREES; ++t) {
        const int idx = leaf_idx[(size_t)t * BATCH + b];
        acc += leaves[((size_t)t * NUM_LEAVES + idx) * NUM_CLASSES + c];
    }
    out[(size_t)b * NUM_CLASSES + c] = acc * (1.0f / NUM_TREES);
}

extern "C" void kernel_launch(void* const* d_in, const int* in_sizes, int n_in,
                              void* d_out, int out_size, void* d_ws, size_t ws_size,
                              hipStream_t stream) {
    const float* x      = (const float*)d_in[0];  // [8192, 256]
    const float* node_w = (const float*)d_in[1];  // [100, 31, 256]
    const float* node_b = (const float*)d_in[2];  // [100, 31]
    const float* leaves = (const float*)d_in[3];  // [100, 32, 64]
    float* out = (float*)d_out;                   // [8192, 64]

    unsigned char* leaf_idx = (unsigned char*)d_ws;  // 100*8192 = 800 KB

    dim3 grid1(BATCH / 512, NUM_TREES);   // 16 x 100 blocks
    forest_scores_kernel<<<grid1, 256, 0, stream>>>(x, node_w, node_b, leaf_idx);

    forest_avg_kernel<<<BATCH / 4, 256, 0, stream>>>(leaves, leaf_idx, out);
}